// WeightGCN_26585847562450
// MI455X (gfx1250) — compile-verified
//
#include <hip/hip_runtime.h>

#define N_USER 27094
#define M_SPOT 42852
#define HIDDEN 64
#define NUM_LAYERS 3

// ---------------------------------------------------------------------------
// Device-scope fp32 atomic add via native CDNA5 instruction.
// scope:SCOPE_DEV => RMW performed at L2 (coherent across all WGPs).
// No-return form => tracked with STOREcnt; s_endpgm does an implicit
// wait-idle, and no kernel reads its own atomic results.
// ---------------------------------------------------------------------------
static __device__ __forceinline__ void gadd_f32(float* p, float v) {
#if defined(__gfx1250__) || defined(__AMDGCN__)
  asm volatile("global_atomic_add_f32 %0, %1, off scope:SCOPE_DEV"
               :
               : "v"(p), "v"(v)
               : "memory");
#else
  atomicAdd(p, v);
#endif
}

// ---------------------------------------------------------------------------
// Elementwise helpers
// ---------------------------------------------------------------------------
__global__ void k_fill(float* __restrict__ p, long long n, float v) {
  long long i = (long long)blockIdx.x * blockDim.x + threadIdx.x;
  long long stride = (long long)gridDim.x * blockDim.x;
  for (; i < n; i += stride) p[i] = v;
}

__global__ void k_copy(float* __restrict__ dst, const float* __restrict__ src,
                       long long n) {
  long long i = (long long)blockIdx.x * blockDim.x + threadIdx.x;
  long long stride = (long long)gridDim.x * blockDim.x;
  for (; i < n; i += stride) dst[i] = src[i];
}

__global__ void k_axpy(float* __restrict__ acc, const float* __restrict__ add,
                       long long n) {
  long long i = (long long)blockIdx.x * blockDim.x + threadIdx.x;
  long long stride = (long long)gridDim.x * blockDim.x;
  for (; i < n; i += stride) acc[i] += add[i];
}

__global__ void k_scale(float* __restrict__ dst, const float* __restrict__ src,
                        long long n, float s) {
  long long i = (long long)blockIdx.x * blockDim.x + threadIdx.x;
  long long stride = (long long)gridDim.x * blockDim.x;
  for (; i < n; i += stride) dst[i] = src[i] * s;
}

// deg[col[e]] += (w ? w[e] : 1)
__global__ void k_deg(const int* __restrict__ col, const float* __restrict__ w,
                      float* __restrict__ deg, int n) {
  int e = blockIdx.x * blockDim.x + threadIdx.x;
  if (e >= n) return;
  float ww = w ? w[e] : 1.0f;
  gadd_f32(deg + col[e], ww);
}

// bipartite degree counts (both sides in one pass)
__global__ void k_count2(const int* __restrict__ uu, const int* __restrict__ ss,
                         float* __restrict__ cu, float* __restrict__ cs, int n) {
  int e = blockIdx.x * blockDim.x + threadIdx.x;
  if (e >= n) return;
  gadd_f32(cu + uu[e], 1.0f);
  gadd_f32(cs + ss[e], 1.0f);
}

// d[i] = d[i] > 0 ? rsqrt(d[i]) : 0   (in place)
__global__ void k_rsqrt(float* __restrict__ d, int n) {
  int i = blockIdx.x * blockDim.x + threadIdx.x;
  if (i >= n) return;
  float x = d[i];
  d[i] = (x > 0.0f) ? rsqrtf(x) : 0.0f;
}

// inv[e] = rsqrt(cu[uu[e]] * cs[ss[e]])  (product >= 1 by construction)
__global__ void k_invdiv(const int* __restrict__ uu, const int* __restrict__ ss,
                         const float* __restrict__ cu, const float* __restrict__ cs,
                         float* __restrict__ inv, int n) {
  int e = blockIdx.x * blockDim.x + threadIdx.x;
  if (e >= n) return;
  inv[e] = rsqrtf(cu[uu[e]] * cs[ss[e]]);
}

// ---------------------------------------------------------------------------
// Light conv edge kernel: out[col[e]] += dis[row]*w*dis[col] * x[row]
// 16 lanes per edge, float4 per lane (global_load_b128 gather, 4 fp32 atomics)
// ---------------------------------------------------------------------------
__global__ void k_lightconv(const float* __restrict__ x,
                            const int* __restrict__ row,
                            const int* __restrict__ col,
                            const float* __restrict__ w,
                            const float* __restrict__ dis,
                            float* __restrict__ out, int n) {
  int t = blockIdx.x * blockDim.x + threadIdx.x;
  int e = t >> 4;
  int lane = t & 15;
  if (e >= n) return;
  if (lane == 0) {
    __builtin_prefetch(row + e + 4096, 0, 1);
    __builtin_prefetch(col + e + 4096, 0, 1);
  }
  int r = row[e];
  int c = col[e];
  float ww = w ? w[e] : 1.0f;
  float norm = dis[r] * ww * dis[c];
  float4 v = ((const float4*)(x + (size_t)r * HIDDEN))[lane];
  float* dst = out + (size_t)c * HIDDEN + (size_t)lane * 4;
  gadd_f32(dst + 0, v.x * norm);
  gadd_f32(dst + 1, v.y * norm);
  gadd_f32(dst + 2, v.z * norm);
  gadd_f32(dst + 3, v.w * norm);
}

// ---------------------------------------------------------------------------
// Bipartite propagation, both directions fused:
//   unew[u] += scur[s] * inv ;  snew[s] += ucur[u] * inv
//
// Tiling: 256-thread block owns 64 edges. The 192 per-edge scalars
// (u idx, s idx, inv) are staged into LDS with GLOBAL_LOAD_ASYNC_TO_LDS_B32
// (one b32 per lane, threads 0..191), completed with s_wait_asynccnt 0 +
// block barrier, then consumed as ds_load broadcasts. Feature rows are
// gathered as float4 (global_load_b128) and scattered with device-scope
// fp32 atomics (4 per lane per direction).
// ---------------------------------------------------------------------------
__global__ void k_prop(const float* __restrict__ ucur,
                       const float* __restrict__ scur,
                       const int* __restrict__ uu,
                       const int* __restrict__ ss,
                       const float* __restrict__ inv,
                       float* __restrict__ unew,
                       float* __restrict__ snew, int n) {
  __shared__ unsigned smem[192];  // [0,64)=u idx  [64,128)=s idx  [128,192)=inv
  const int tid = threadIdx.x;
  const long long e_base = (long long)blockIdx.x * 64;

  if (e_base + 64 <= (long long)n) {
    // Full tile: async global->LDS staging (ASYNCcnt domain).
    if (tid < 192) {
      const int slot = tid & 63;
      const unsigned* g =
          (tid < 64)  ? (const unsigned*)(uu + e_base + slot)
        : (tid < 128) ? (const unsigned*)(ss + e_base + slot)
                      : (const unsigned*)(inv + e_base + slot);
      // Generic LDS address: low 32 bits are the wave-relative LDS offset.
      unsigned lds_off = (unsigned)(size_t)(&smem[tid]);
      asm volatile("global_load_async_to_lds_b32 %0, %1, off"
                   :
                   : "v"(lds_off), "v"(g)
                   : "memory");
    }
    asm volatile("s_wait_asynccnt 0x0" ::: "memory");
  } else {
    // Tail tile (block-uniform branch): plain loads into LDS.
    if (tid < 192) {
      const int slot = tid & 63;
      const long long e = e_base + slot;
      if (e < (long long)n) {
        smem[tid] = (tid < 64)  ? ((const unsigned*)uu)[e]
                  : (tid < 128) ? ((const unsigned*)ss)[e]
                                : ((const unsigned*)inv)[e];
      }
    }
  }
  __syncthreads();

  const int lane = tid & 15;   // float4 chunk within the 64-dim row
  const int grp  = tid >> 4;   // 0..15 edge-slot group
#pragma unroll
  for (int it = 0; it < 4; ++it) {
    const int slot = it * 16 + grp;          // 0..63
    const long long e = e_base + slot;
    if (e >= (long long)n) continue;
    const int u  = (int)smem[slot];
    const int s  = (int)smem[64 + slot];
    const float iv = __uint_as_float(smem[128 + slot]);
    float4 sv = ((const float4*)(scur + (size_t)s * HIDDEN))[lane];
    float4 uv = ((const float4*)(ucur + (size_t)u * HIDDEN))[lane];
    float* du = unew + (size_t)u * HIDDEN + (size_t)lane * 4;
    float* ds = snew + (size_t)s * HIDDEN + (size_t)lane * 4;
    gadd_f32(du + 0, sv.x * iv);
    gadd_f32(du + 1, sv.y * iv);
    gadd_f32(du + 2, sv.z * iv);
    gadd_f32(du + 3, sv.w * iv);
    gadd_f32(ds + 0, uv.x * iv);
    gadd_f32(ds + 1, uv.y * iv);
    gadd_f32(ds + 2, uv.z * iv);
    gadd_f32(ds + 3, uv.w * iv);
  }
}

// ---------------------------------------------------------------------------
// Host launcher
// ---------------------------------------------------------------------------
extern "C" void kernel_launch(void* const* d_in, const int* in_sizes, int n_in,
                              void* d_out, int out_size, void* d_ws, size_t ws_size,
                              hipStream_t stream) {
  const float* user_emb = (const float*)d_in[0];
  const float* spot_emb = (const float*)d_in[1];
  const float* spot_w   = (const float*)d_in[2];
  const int*   us_user  = (const int*)d_in[3];
  const int*   us_spot  = (const int*)d_in[4];
  const int*   ue       = (const int*)d_in[5];
  const int*   se       = (const int*)d_in[6];

  const int n_us = in_sizes[3];
  const int n_ue = in_sizes[5] / 2;
  const int n_se = in_sizes[6] / 2;

  const int* ue_row = ue;           // user_edge_index[0]
  const int* ue_col = ue + n_ue;    // user_edge_index[1]
  const int* se_row = se;
  const int* se_col = se + n_se;

  // ---- workspace layout (all regions 16B aligned) ----
  float* ws = (float*)d_ws;
  auto pad4 = [](long long x) { return (x + 3LL) & ~3LL; };
  long long off = 0;
  float* deg_u = ws + off; off += pad4(N_USER);   // user-graph deg -> dis
  float* deg_s = ws + off; off += pad4(M_SPOT);   // spot-graph deg -> dis
  float* cu    = ws + off; off += pad4(N_USER);   // bipartite user counts
  float* cs    = ws + off; off += pad4(M_SPOT);   // bipartite spot counts
  const long long zero_span = off;                // zero these four together
  float* inv   = ws + off; off += pad4(n_us);
  const long long un = (long long)N_USER * HIDDEN;
  const long long sn = (long long)M_SPOT * HIDDEN;
  float* ubuf0 = ws + off; off += un;
  float* ubuf1 = ws + off; off += un;
  float* uacc  = ws + off; off += un;
  float* sbuf0 = ws + off; off += sn;
  float* sbuf1 = ws + off; off += sn;
  float* sacc  = ws + off; off += sn;
  (void)ws_size; (void)n_in; (void)out_size;

  const int B = 256;
  auto nb = [&](long long work) { return (unsigned)((work + B - 1) / B); };

  // Phase A: degrees
  k_fill<<<nb(zero_span), B, 0, stream>>>(deg_u, zero_span, 0.0f);
  k_deg<<<nb(n_ue), B, 0, stream>>>(ue_col, nullptr, deg_u, n_ue);
  k_deg<<<nb(n_se), B, 0, stream>>>(se_col, spot_w, deg_s, n_se);
  k_count2<<<nb(n_us), B, 0, stream>>>(us_user, us_spot, cu, cs, n_us);
  k_rsqrt<<<nb(N_USER), B, 0, stream>>>(deg_u, N_USER);   // -> dis_u
  k_rsqrt<<<nb(M_SPOT), B, 0, stream>>>(deg_s, M_SPOT);   // -> dis_s
  k_invdiv<<<nb(n_us), B, 0, stream>>>(us_user, us_spot, cu, cs, inv, n_us);

  // Phase B: light conv (x + conv(x)); gather from original emb, add into cur
  k_copy<<<nb(un), B, 0, stream>>>(ubuf0, user_emb, un);
  k_copy<<<nb(sn), B, 0, stream>>>(sbuf0, spot_emb, sn);
  k_lightconv<<<nb((long long)n_ue * 16), B, 0, stream>>>(
      user_emb, ue_row, ue_col, nullptr, deg_u, ubuf0, n_ue);
  k_lightconv<<<nb((long long)n_se * 16), B, 0, stream>>>(
      spot_emb, se_row, se_col, spot_w, deg_s, sbuf0, n_se);

  // Accumulators start at layer-0 features
  k_copy<<<nb(un), B, 0, stream>>>(uacc, ubuf0, un);
  k_copy<<<nb(sn), B, 0, stream>>>(sacc, sbuf0, sn);

  // Phase C: 3 bipartite propagation layers (ping-pong)
  float* ucur = ubuf0; float* unew = ubuf1;
  float* scur = sbuf0; float* snew = sbuf1;
  const unsigned prop_blocks = (unsigned)(((long long)n_us + 63) / 64);
  for (int l = 0; l < NUM_LAYERS; ++l) {
    k_fill<<<nb(un), B, 0, stream>>>(unew, un, 0.0f);
    k_fill<<<nb(sn), B, 0, stream>>>(snew, sn, 0.0f);
    k_prop<<<prop_blocks, B, 0, stream>>>(
        ucur, scur, us_user, us_spot, inv, unew, snew, n_us);
    k_axpy<<<nb(un), B, 0, stream>>>(uacc, unew, un);
    k_axpy<<<nb(sn), B, 0, stream>>>(sacc, snew, sn);
    float* t;
    t = ucur; ucur = unew; unew = t;
    t = scur; scur = snew; snew = t;
  }

  // Phase D: output = concat(spot_acc, user_acc) * 1/(L+1)
  float* out = (float*)d_out;
  const float scale = 1.0f / (float)(NUM_LAYERS + 1);
  k_scale<<<nb(sn), B, 0, stream>>>(out, sacc, sn, scale);
  k_scale<<<nb(un), B, 0, stream>>>(out + sn, uacc, un, scale);
}